// BaseModel_83777632076450
// MI455X (gfx1250) — compile-verified
//
#include <hip/hip_runtime.h>

// Problem constants (match reference)
#define B_   16
#define C_   384
#define N_   16384
#define M_   256
#define P_   128
#define ROWS (B_ * M_)          // 4096 pooled rows
#define EPS_ 1e-5f

// Pooling scatter tiling: 64 channels per LDS tile, channel-major
// (tile[c*256 + lab] -> LDS bank = lab & 63, random across lanes).
// 6 channel chunks, 8 N chunks -> 768 workgroups streaming features once.
#define CC   64
#define NCH  8

typedef __attribute__((ext_vector_type(2))) float v2f;
typedef __attribute__((ext_vector_type(8))) float v8f;

// ---------------------------------------------------------------------------
__global__ void zero_kernel(float* __restrict__ p, int n) {
    int i = blockIdx.x * blockDim.x + threadIdx.x;
    if (i < n) p[i] = 0.0f;
}

// ---------------------------------------------------------------------------
// Naive transpose: Wt[c, k] = W[k, c].  (weights are tiny; one-time cost)
__global__ void transpose_kernel(const float* __restrict__ Win,
                                 float* __restrict__ Wt, int K, int Ncols) {
    int i = blockIdx.x * blockDim.x + threadIdx.x;
    if (i < K * Ncols) {
        int k = i / Ncols, c = i % Ncols;
        Wt[(size_t)c * K + k] = Win[i];
    }
}

// ---------------------------------------------------------------------------
// Per-batch label histogram -> counts (float). blockDim == M_ == 256.
__global__ void counts_kernel(const int* __restrict__ labels,
                              float* __restrict__ counts) {
    __shared__ unsigned hist[M_];
    const int b = blockIdx.x;
    hist[threadIdx.x] = 0u;
    __syncthreads();
    for (int n = threadIdx.x; n < N_; n += M_)
        atomicAdd(&hist[labels[b * N_ + n]], 1u);          // ds_add_u32
    __syncthreads();
    counts[b * M_ + threadIdx.x] = (float)hist[threadIdx.x];
}

// ---------------------------------------------------------------------------
// Segment-sum pooling with LDS privatization (the 384 MB HBM-bound pass).
// Grid = B_ * (C_/CC) * NCH blocks of 256 threads.  float4 feature stream,
// channel-major LDS tile [CC][M_] to avoid systematic bank conflicts,
// global_atomic_add_f32 flush.
__global__ void scatter_kernel(const float* __restrict__ feat,
                               const int*   __restrict__ labels,
                               float*       __restrict__ sums) {
    __shared__ float tile[CC * M_];                        // 64 KB, [c][m]
    const int bid  = blockIdx.x;
    const int nch  = bid % NCH;
    const int t2   = bid / NCH;
    const int cch  = t2 % (C_ / CC);
    const int b    = t2 / (C_ / CC);
    const int cbase = cch * CC;
    const int nbase = nch * (N_ / NCH);

    for (int i = threadIdx.x; i < CC * M_; i += blockDim.x) tile[i] = 0.0f;
    __syncthreads();

    // each k-step: 256 threads x float4 = 1024 points
    for (int k = 0; k < N_ / NCH; k += 1024) {
        const int n = nbase + k + 4 * threadIdx.x;
        const int4 lab = *(const int4*)&labels[b * N_ + n];
        const float* fp = feat + ((size_t)b * C_ + cbase) * N_ + n;
#pragma unroll 4
        for (int c = 0; c < CC; ++c) {
            const float4 v = *(const float4*)(fp + (size_t)c * N_);
            float* row = &tile[c * M_];
            atomicAdd(&row[lab.x], v.x);                   // ds_add_f32
            atomicAdd(&row[lab.y], v.y);
            atomicAdd(&row[lab.z], v.z);
            atomicAdd(&row[lab.w], v.w);
        }
    }
    __syncthreads();

    // flush: lane -> m (conflict-free LDS reads), loop c
    const int m = threadIdx.x;
    for (int c = 0; c < CC; ++c)
        atomicAdd(&sums[((size_t)b * M_ + m) * C_ + cbase + c], tile[c * M_ + m]);
}

// ---------------------------------------------------------------------------
// Fused GEMM using V_WMMA_F32_16X16X4_F32. One wave computes one 16x16 output
// tile, K in steps of 4 (96 WMMA issues for K=384).  Wt is [Ncols, K]
// (pre-transposed) so both fragments are contiguous b64 loads.
//   MODE 0: A := sums / counts (segment mean),  out = A@W + bias
//   MODE 1: A := relu(A*scale + shift),         out = A@W + bias
//   MODE 2: A := relu(A*scale + shift),         out = A@W       (final proj)
template <int MODE>
__global__ void gemm_wmma_kernel(const float* __restrict__ A,
                                 const float* __restrict__ Wt,
                                 const float* __restrict__ bias,
                                 const float* __restrict__ counts,
                                 const float* __restrict__ scale,
                                 const float* __restrict__ shift,
                                 float* __restrict__ out,
                                 int K, int Ncols, int n_tiles) {
    const int gtid   = blockIdx.x * blockDim.x + threadIdx.x;
    const int wave   = gtid >> 5;          // wave32
    const int lane   = threadIdx.x & 31;
    const int half   = lane >> 4;
    const int lane16 = lane & 15;
    const int tile_m = wave / n_tiles;
    const int tile_n = wave % n_tiles;
    const int arow   = tile_m * 16 + lane16;   // A-matrix row for this lane
    const int col    = tile_n * 16 + lane16;   // B/C/D column for this lane

    float inv = 1.0f;
    if (MODE == 0) {
        const float c = counts[arow];
        inv = (c != 0.0f) ? (1.0f / c) : 1.0f;
    }
    // element ka = k0 + 2*half  ->  float2 index k0/2 + half
    const float2* ap = (const float2*)(A  + (size_t)arow * K) + half;
    const float2* bp = (const float2*)(Wt + (size_t)col  * K) + half;
    const float2* sc = (const float2*)scale + half;
    const float2* sh = (const float2*)shift + half;

    v8f acc = {0.f, 0.f, 0.f, 0.f, 0.f, 0.f, 0.f, 0.f};
#pragma unroll 4
    for (int k2 = 0; k2 < K / 2; k2 += 2) {    // k0 = 2*k2
        float2 av = ap[k2];
        if (MODE == 0) {
            av.x *= inv;  av.y *= inv;
        } else {
            const float2 s = sc[k2];
            const float2 t = sh[k2];
            av.x = fmaxf(av.x * s.x + t.x, 0.0f);
            av.y = fmaxf(av.y * s.y + t.y, 0.0f);
        }
        const float2 bv = bp[k2];
        v2f a;  a.x = av.x;  a.y = av.y;
        v2f bb; bb.x = bv.x; bb.y = bv.y;
        // (neg_a, A, neg_b, B, c_mod, C, reuse_a, reuse_b)
        acc = __builtin_amdgcn_wmma_f32_16x16x4_f32(false, a, false, bb,
                                                    (short)0, acc, false, false);
    }

    const float bv = (MODE == 2) ? 0.0f : bias[col];
#pragma unroll
    for (int i = 0; i < 8; ++i) {
        const int r = tile_m * 16 + half * 8 + i;   // C/D layout: vgpr i -> row i+8*half
        out[(size_t)r * Ncols + col] = acc[i] + bv;
    }
}

// ---------------------------------------------------------------------------
// Per-column batch stats over ROWS rows, folded into BN affine:
// scale = gamma * rsqrt(var+eps);  shift = beta - mean*scale.  Grid = Ncols.
__global__ void stats_kernel(const float* __restrict__ X,
                             const float* __restrict__ gamma,
                             const float* __restrict__ beta,
                             float* __restrict__ scale,
                             float* __restrict__ shift,
                             int Ncols) {
    __shared__ float s1[256];
    __shared__ float s2[256];
    const int j = blockIdx.x;
    float s = 0.0f, q = 0.0f;
    for (int r = threadIdx.x; r < ROWS; r += 256) {
        const float v = X[(size_t)r * Ncols + j];
        s += v; q += v * v;
    }
    s1[threadIdx.x] = s; s2[threadIdx.x] = q;
    __syncthreads();
    for (int off = 128; off > 0; off >>= 1) {
        if (threadIdx.x < off) {
            s1[threadIdx.x] += s1[threadIdx.x + off];
            s2[threadIdx.x] += s2[threadIdx.x + off];
        }
        __syncthreads();
    }
    if (threadIdx.x == 0) {
        const float mu  = s1[0] * (1.0f / (float)ROWS);
        const float var = s2[0] * (1.0f / (float)ROWS) - mu * mu;   // biased (var(0))
        const float sc  = gamma[j] * rsqrtf(var + EPS_);
        scale[j] = sc;
        shift[j] = beta[j] - mu * sc;
    }
}

// ---------------------------------------------------------------------------
// samples_idx = repeat(arange(16), 256); patch_idx = tile(arange(256), 16)
__global__ void idx_kernel(int* __restrict__ out) {
    const int i = blockIdx.x * blockDim.x + threadIdx.x;   // ROWS total
    out[i]        = i >> 8;        // i / 256
    out[ROWS + i] = i & (M_ - 1);  // i % 256
}

// ---------------------------------------------------------------------------
extern "C" void kernel_launch(void* const* d_in, const int* in_sizes, int n_in,
                              void* d_out, int out_size, void* d_ws, size_t ws_size,
                              hipStream_t stream) {
    const float* feat   = (const float*)d_in[0];
    const int*   labels = (const int*)  d_in[1];
    const float* W1  = (const float*)d_in[2];
    const float* b1  = (const float*)d_in[3];
    const float* g1  = (const float*)d_in[4];
    const float* be1 = (const float*)d_in[5];
    const float* W2  = (const float*)d_in[6];
    const float* b2  = (const float*)d_in[7];
    const float* g2  = (const float*)d_in[8];
    const float* be2 = (const float*)d_in[9];
    const float* W3  = (const float*)d_in[10];

    // Workspace layout (fp32): ~20.3 MB total
    float* ws     = (float*)d_ws;
    float* sums   = ws;                           // ROWS*C_
    float* counts = sums + (size_t)ROWS * C_;     // ROWS
    float* X1     = counts + ROWS;                // ROWS*C_ (pre-BN layer1)
    float* X2     = X1 + (size_t)ROWS * C_;       // ROWS*C_ (pre-BN layer2)
    float* Wt1    = X2 + (size_t)ROWS * C_;       // C_*C_  (W1^T)
    float* Wt2    = Wt1 + (size_t)C_ * C_;        // C_*C_  (W2^T)
    float* Wt3    = Wt2 + (size_t)C_ * C_;        // C_*P_  (W3^T)
    float* scale1 = Wt3 + (size_t)C_ * P_;        // C_
    float* shift1 = scale1 + C_;                  // C_
    float* scale2 = shift1 + C_;                  // C_
    float* shift2 = scale2 + C_;                  // C_

    // 0) pre-transpose weights (B fragments become contiguous b64 loads)
    transpose_kernel<<<(C_ * C_ + 255) / 256, 256, 0, stream>>>(W1, Wt1, C_, C_);
    transpose_kernel<<<(C_ * C_ + 255) / 256, 256, 0, stream>>>(W2, Wt2, C_, C_);
    transpose_kernel<<<(C_ * P_ + 255) / 256, 256, 0, stream>>>(W3, Wt3, C_, P_);

    // 1) zero segment-sum accumulators (+counts)
    const int nz = ROWS * C_ + ROWS;
    zero_kernel<<<(nz + 255) / 256, 256, 0, stream>>>(sums, nz);

    // 2) per-batch label histogram
    counts_kernel<<<B_, M_, 0, stream>>>(labels, counts);

    // 3) streaming segment-sum
    scatter_kernel<<<B_ * (C_ / CC) * NCH, 256, 0, stream>>>(feat, labels, sums);

    // 4) layer 1: X1 = (sums/counts) @ W1 + b1
    gemm_wmma_kernel<0><<<(ROWS / 16) * (C_ / 16) / 8, 256, 0, stream>>>(
        sums, Wt1, b1, counts, nullptr, nullptr, X1, C_, C_, C_ / 16);

    // 5) batch stats of X1 -> folded BN affine
    stats_kernel<<<C_, 256, 0, stream>>>(X1, g1, be1, scale1, shift1, C_);

    // 6) layer 2: X2 = relu(bn(X1)) @ W2 + b2   (BN+ReLU fused into A load)
    gemm_wmma_kernel<1><<<(ROWS / 16) * (C_ / 16) / 8, 256, 0, stream>>>(
        X1, Wt2, b2, nullptr, scale1, shift1, X2, C_, C_, C_ / 16);

    // 7) batch stats of X2
    stats_kernel<<<C_, 256, 0, stream>>>(X2, g2, be2, scale2, shift2, C_);

    // 8) projection: embs = relu(bn(X2)) @ W3  -> d_out head
    float* embs = (float*)d_out;
    gemm_wmma_kernel<2><<<(ROWS / 16) * (P_ / 16) / 8, 256, 0, stream>>>(
        X2, Wt3, nullptr, nullptr, scale2, shift2, embs, C_, P_, P_ / 16);

    // 9) index outputs (int32 bit patterns) appended after embs
    int* idx_out = (int*)(embs + (size_t)ROWS * P_);
    idx_kernel<<<ROWS / 256, 256, 0, stream>>>(idx_out);
}